// Two_Headed_Loss_39814346834260
// MI455X (gfx1250) — compile-verified
//
#include <hip/hip_runtime.h>
#include <hip/hip_bf16.h>
#include <math.h>

typedef __attribute__((ext_vector_type(2))) float v2f;
typedef __attribute__((ext_vector_type(8))) float v8f;

#ifndef LM_BLOCK
#define LM_BLOCK 256
#endif

// ---------------------------------------------------------------------------
// Kernel 1: per-row online logsumexp + NLL for the LM head.
// One block per row (N=4096 blocks). 500 MB streamed exactly once.
// ---------------------------------------------------------------------------
__global__ __launch_bounds__(LM_BLOCK) void lm_ce_kernel(
    const float* __restrict__ logits,   // [N, V]
    const int*   __restrict__ labels,   // [N]
    float* __restrict__ nll_out,        // [N]
    float* __restrict__ valid_out,      // [N]
    int V) {
  const int n   = blockIdx.x;
  const int tid = threadIdx.x;
  const float*  row  = logits + (size_t)n * (size_t)V;
  const float2* row2 = (const float2*)row;   // rows are 8B aligned (V even typ.)
  const int V2 = V >> 1;

  // thread-local online logsumexp
  float m = -INFINITY;
  float s = 0.0f;
  for (int i = tid; i < V2; i += LM_BLOCK) {
    // stream-prefetch well ahead (speculative; OOB silently dropped)
    __builtin_prefetch(&row2[i + 8 * LM_BLOCK], 0, 1);
    float2 v = row2[i];
    // element 0
    if (v.x > m) { s = s * __expf(m - v.x) + 1.0f; m = v.x; }
    else         { s += __expf(v.x - m); }
    // element 1
    if (v.y > m) { s = s * __expf(m - v.y) + 1.0f; m = v.y; }
    else         { s += __expf(v.y - m); }
  }
  if ((V & 1) && tid == 0) {
    float v = row[V - 1];
    if (v > m) { s = s * __expf(m - v) + 1.0f; m = v; }
    else       { s += __expf(v - m); }
  }

  // block tree reduction of (m, s) pairs — fixed order, deterministic
  __shared__ float sm[LM_BLOCK];
  __shared__ float ss[LM_BLOCK];
  sm[tid] = m; ss[tid] = s;
  __syncthreads();
  for (int off = LM_BLOCK >> 1; off > 0; off >>= 1) {
    if (tid < off) {
      float m1 = sm[tid],       s1 = ss[tid];
      float m2 = sm[tid + off], s2 = ss[tid + off];
      float M  = fmaxf(m1, m2);
      sm[tid] = M;
      ss[tid] = s1 * __expf(m1 - M) + s2 * __expf(m2 - M);
    }
    __syncthreads();
  }

  if (tid == 0) {
    int lbl = labels[n];
    if (lbl >= 0) {
      float lse = sm[0] + __logf(ss[0]);
      nll_out[n]   = lse - row[lbl];
      valid_out[n] = 1.0f;
    } else {
      nll_out[n]   = 0.0f;
      valid_out[n] = 0.0f;
    }
  }
}

// ---------------------------------------------------------------------------
// Kernel 2: blank head.  G = X X^T via V_WMMA_F32_16X16X4_F32, one wave per
// 16x16 tile; mask + softplus applied on the accumulator fragment in-regs.
// ---------------------------------------------------------------------------
__device__ __forceinline__ float softplus_f(float x) {
  // numerically stable: max(x,0) + log1p(exp(-|x|))
  return fmaxf(x, 0.0f) + log1pf(__expf(-fabsf(x)));
}

__global__ __launch_bounds__(256) void blank_gram_kernel(
    const float* __restrict__ X,        // [B, D]
    const int*   __restrict__ labels,   // [B]
    float* __restrict__ tile_sum,       // [ntiles]
    float* __restrict__ tile_cnt,       // [ntiles]
    int D, int tiles_per_side) {
  const int lane = threadIdx.x & 31;
  const int wave = threadIdx.x >> 5;
  const int tile = blockIdx.x * (blockDim.x >> 5) + wave;
  const int ti = tile / tiles_per_side;
  const int tj = tile % tiles_per_side;

  const int half = lane >> 4;     // 0: K=k..k+1, 1: K=k+2..k+3
  const int r    = lane & 15;     // row within tile (A) / col within tile (B)
  const int D2   = D >> 1;

  const float2* X2 = (const float2*)X;
  const size_t baseA = (size_t)(ti * 16 + r) * (size_t)D2;
  const size_t baseB = (size_t)(tj * 16 + r) * (size_t)D2;   // B = X^T: cols are rows of X

  v8f c = {};   // 16x16 f32 accumulator fragment (8 VGPRs)
  for (int k2 = 0; k2 < D2; k2 += 2) {       // K advances by 4 per WMMA
    float2 av = X2[baseA + k2 + half];
    float2 bv = X2[baseB + k2 + half];
    v2f a; a.x = av.x; a.y = av.y;
    v2f b; b.x = bv.x; b.y = bv.y;
    // D = A(16x4) * B(4x16) + C, f32
    c = __builtin_amdgcn_wmma_f32_16x16x4_f32(
        /*neg_a=*/false, a, /*neg_b=*/false, b,
        /*c_mod=*/(short)0, c, /*reuse_a=*/false, /*reuse_b=*/false);
  }

  // C/D layout: VGPR rr, lane l -> i = 16*ti + rr + 8*(l>=16), j = 16*tj + (l&15)
  const int j  = tj * 16 + r;
  const int yj = (labels[j] == 1);
  float sum = 0.0f, cnt = 0.0f;
#pragma unroll
  for (int rr = 0; rr < 8; ++rr) {
    const int i  = ti * 16 + rr + 8 * half;
    const int yi = (labels[i] == 1);
    const float g = c[rr];
    if (yi && yj && (i < j)) { sum += softplus_f(-g); cnt += 1.0f; }  // pos pair, i<j
    if (yi && !yj)           { sum += softplus_f( g); cnt += 1.0f; }  // pos->neg ordered
  }

  // wave32 butterfly reduction
#pragma unroll
  for (int off = 16; off > 0; off >>= 1) {
    sum += __shfl_xor(sum, off, 32);
    cnt += __shfl_xor(cnt, off, 32);
  }
  if (lane == 0) {
    tile_sum[tile] = sum;
    tile_cnt[tile] = cnt;
  }
}

// ---------------------------------------------------------------------------
// Kernel 3: deterministic single-block finalize -> scalar loss.
// ---------------------------------------------------------------------------
__global__ __launch_bounds__(256) void finalize_kernel(
    const float* __restrict__ nll,      // [N]
    const float* __restrict__ valid,    // [N]
    const float* __restrict__ tsum,     // [ntiles]
    const float* __restrict__ tcnt,     // [ntiles]
    float* __restrict__ out, int N, int ntiles) {
  __shared__ float s0[256], s1[256], s2[256], s3[256];
  const int tid = threadIdx.x;
  float a = 0.0f, b = 0.0f, cc = 0.0f, d = 0.0f;
  for (int i = tid; i < N; i += 256)      { a += nll[i];  b += valid[i]; }
  for (int i = tid; i < ntiles; i += 256) { cc += tsum[i]; d += tcnt[i]; }
  s0[tid] = a; s1[tid] = b; s2[tid] = cc; s3[tid] = d;
  __syncthreads();
  for (int off = 128; off > 0; off >>= 1) {
    if (tid < off) {
      s0[tid] += s0[tid + off];
      s1[tid] += s1[tid + off];
      s2[tid] += s2[tid + off];
      s3[tid] += s3[tid + off];
    }
    __syncthreads();
  }
  if (tid == 0) {
    float vcount  = fmaxf(s1[0], 1.0f);
    float lm_loss = s0[0] / vcount;
    float npairs  = fmaxf(s3[0], 1.0f);
    float bl_loss = s2[0] / npairs;
    out[0] = lm_loss + bl_loss;
  }
}

// ---------------------------------------------------------------------------
extern "C" void kernel_launch(void* const* d_in, const int* in_sizes, int n_in,
                              void* d_out, int out_size, void* d_ws, size_t ws_size,
                              hipStream_t stream) {
  const float* lm_logits    = (const float*)d_in[0];   // [N, V]
  const float* blank_logits = (const float*)d_in[1];   // [B, D]
  const int*   lm_labels    = (const int*)d_in[2];     // [N]
  const int*   blank_labels = (const int*)d_in[3];     // [B]

  const int N = in_sizes[2];
  const int V = in_sizes[0] / N;
  const int B = in_sizes[3];
  const int D = in_sizes[1] / B;

  const int tiles_per_side = B / 16;               // 16
  const int ntiles = tiles_per_side * tiles_per_side; // 256

  float* ws       = (float*)d_ws;
  float* ws_nll   = ws;                 // [N]
  float* ws_valid = ws + N;             // [N]
  float* ws_tsum  = ws + 2 * N;         // [ntiles]
  float* ws_tcnt  = ws + 2 * N + ntiles;// [ntiles]

  // Stage 1: LM head per-row NLL (memory-bound, ~500 MB streamed once)
  lm_ce_kernel<<<N, LM_BLOCK, 0, stream>>>(lm_logits, lm_labels,
                                           ws_nll, ws_valid, V);

  // Stage 2: blank head Gram + masked softplus via WMMA (8 waves/block)
  const int waves_per_block = 8;
  const int blocks2 = ntiles / waves_per_block;    // 32
  blank_gram_kernel<<<blocks2, 32 * waves_per_block, 0, stream>>>(
      blank_logits, blank_labels, ws_tsum, ws_tcnt, D, tiles_per_side);

  // Stage 3: deterministic finalize
  finalize_kernel<<<1, 256, 0, stream>>>(ws_nll, ws_valid, ws_tsum, ws_tcnt,
                                         (float*)d_out, N, ntiles);
}